// GaussianMomentDescriptor_3015067041843
// MI455X (gfx1250) — compile-verified
//
#include <hip/hip_runtime.h>
#include <hip/hip_bf16.h>
#include <math.h>

#define N_RADIAL 5
#define N_BASIS  7
#define N_SPEC   119
#define NCOMP    100   // unique symmetric moment components per atom: 5 + 15 + 30 + 50
#define NFEAT    360

typedef float v2f __attribute__((ext_vector_type(2)));
typedef float v8f __attribute__((ext_vector_type(8)));

// ---- symmetric-tensor unique-index maps ------------------------------------
__device__ __host__ __forceinline__ constexpr int s2i(int i, int j) {  // 6 unique
    int a = i < j ? i : j, b = i < j ? j : i;
    return a + (b * (b + 1)) / 2;
}
__device__ __host__ __forceinline__ constexpr int s3i(int i, int j, int k) { // 10 unique
    int a = i, b = j, c = k, t;
    if (a > b) { t = a; a = b; b = t; }
    if (b > c) { t = b; b = c; c = t; }
    if (a > b) { t = a; a = b; b = t; }
    return a + (b * (b + 1)) / 2 + (c * (c + 1) * (c + 2)) / 6;
}
__device__ __host__ __forceinline__ constexpr int u2map(int k) {
    int i = k / 3, j = k - 3 * i; return s2i(i, j);
}
__device__ __host__ __forceinline__ constexpr int u3map(int k) {
    int i = k / 9; int r = k - 9 * i; int j = r / 3; int kk = r - 3 * j;
    return s3i(i, j, kk);
}

// ---- stage 1: per-edge moments, atomically scattered into per-atom buffer --
__global__ void gm_edge_kernel(const float* __restrict__ R, const int* __restrict__ Z,
                               const int* __restrict__ idx, const float* __restrict__ emb,
                               float* __restrict__ M, int nEdges)
{
    int e = blockIdx.x * blockDim.x + threadIdx.x;
    if (e >= nEdges) return;
    int ai = idx[e];
    int aj = idx[nEdges + e];

    float xi = R[3 * ai + 0], yi = R[3 * ai + 1], zi = R[3 * ai + 2];
    float dx = R[3 * aj + 0] - xi, dy = R[3 * aj + 1] - yi, dz = R[3 * aj + 2] - zi;
    float dr2 = dx * dx + dy * dy + dz * dz;
    float dr  = sqrtf(dr2);
    float inv = 1.0f / dr;
    float d1[3] = { dx * inv, dy * inv, dz * inv };

    const float RMAXF = 6.0f, PIF = 3.14159265358979f;
    float drc = fminf(dr, RMAXF);
    float cut = 0.5f * (cosf(PIF * drc * (1.0f / RMAXF)) + 1.0f);

    const float BETTA  = 1.36111111f;      // (N_BASIS/R_MAX)^2 = 49/36
    const float SCALE0 = 0.36466509f;      // (2*betta/pi)^0.25 / sqrt(7)
    float scale = SCALE0 * cut;

    float basis[N_BASIS];
#pragma unroll
    for (int b = 0; b < N_BASIS; ++b) {
        float sh = 0.5f + 0.785714286f * (float)b;   // R_MIN + (R_MAX-R_MIN)/N_BASIS * b
        float t  = sh - dr;
        basis[b] = expf(-BETTA * t * t);
    }

    int zi_ = Z[ai], zj_ = Z[aj];
    const float* cp = emb + ((size_t)zj_ * N_SPEC + (size_t)zi_) * (N_RADIAL * N_BASIS);
    float rad[N_RADIAL];
#pragma unroll
    for (int r = 0; r < N_RADIAL; ++r) {
        float s = 0.0f;
#pragma unroll
        for (int b = 0; b < N_BASIS; ++b) s += cp[r * N_BASIS + b] * basis[b];
        rad[r] = s * scale;
    }

    float d2[6], d3[10];
#pragma unroll
    for (int i = 0; i < 3; ++i)
#pragma unroll
        for (int j = i; j < 3; ++j) d2[s2i(i, j)] = d1[i] * d1[j];
#pragma unroll
    for (int i = 0; i < 3; ++i)
#pragma unroll
        for (int j = i; j < 3; ++j)
#pragma unroll
            for (int k = j; k < 3; ++k) d3[s3i(i, j, k)] = d2[s2i(i, j)] * d1[k];

    float* base = M + (size_t)aj * NCOMP;
#pragma unroll
    for (int r = 0; r < N_RADIAL; ++r) {
        float v = rad[r];
        unsafeAtomicAdd(base + r, v);
#pragma unroll
        for (int i = 0; i < 3;  ++i) unsafeAtomicAdd(base + 5  + r * 3  + i, v * d1[i]);
#pragma unroll
        for (int u = 0; u < 6;  ++u) unsafeAtomicAdd(base + 20 + r * 6  + u, v * d2[u]);
#pragma unroll
        for (int u = 0; u < 10; ++u) unsafeAtomicAdd(base + 50 + r * 10 + u, v * d3[u]);
    }
}

// ---- stage 2a: per-atom scalar contractions (c0, c4..c7) -------------------
__global__ void gm_atom_kernel(const float* __restrict__ M, float* __restrict__ out, int nAtoms)
{
    int a = blockIdx.x * blockDim.x + threadIdx.x;
    if (a >= nAtoms) return;
    const float* m = M + (size_t)a * NCOMP;

    float M0[5], M1[5][3], M2[5][6], M3[5][10];
#pragma unroll
    for (int r = 0; r < 5; ++r) {
        M0[r] = m[r];
#pragma unroll
        for (int i = 0; i < 3;  ++i) M1[r][i] = m[5  + r * 3  + i];
#pragma unroll
        for (int u = 0; u < 6;  ++u) M2[r][u] = m[20 + r * 6  + u];
#pragma unroll
        for (int u = 0; u < 10; ++u) M3[r][u] = m[50 + r * 10 + u];
    }
    float* o = out + (size_t)a * NFEAT;
#pragma unroll
    for (int r = 0; r < 5; ++r) o[r] = M0[r];   // c0

    // c4 (35 feats @ 50): sum_ijk M2[r,ij] M2[s,ik] M2[t,jk], t3 order (r>=s>=t)
    {
        int p = 0;
#pragma unroll
        for (int r = 0; r < 5; ++r) {
#pragma unroll
            for (int s = 0; s <= r; ++s) {
                float B[3][3];
#pragma unroll
                for (int j = 0; j < 3; ++j)
#pragma unroll
                    for (int k = 0; k < 3; ++k) {
                        float acc = 0.0f;
#pragma unroll
                        for (int i = 0; i < 3; ++i) acc += M2[r][s2i(i, j)] * M2[s][s2i(i, k)];
                        B[j][k] = acc;
                    }
#pragma unroll
                for (int t = 0; t <= s; ++t) {
                    float acc = 0.0f;
#pragma unroll
                    for (int j = 0; j < 3; ++j)
#pragma unroll
                        for (int k = 0; k < 3; ++k) acc += B[j][k] * M2[t][s2i(j, k)];
                    o[50 + p] = acc; ++p;
                }
            }
        }
    }
    // c5 (75 feats @ 85): sum_ij M1[r,i] M1[s,j] M2[t,ij], pairs (r>=s), all t
#pragma unroll
    for (int r = 0; r < 5; ++r)
#pragma unroll
        for (int s = 0; s <= r; ++s) {
            int pair = r * (r + 1) / 2 + s;
#pragma unroll
            for (int t = 0; t < 5; ++t) {
                float acc = 0.0f;
#pragma unroll
                for (int i = 0; i < 3; ++i)
#pragma unroll
                    for (int j = 0; j < 3; ++j) acc += M1[r][i] * M1[s][j] * M2[t][s2i(i, j)];
                o[85 + pair * 5 + t] = acc;
            }
        }
    // c6 (75 feats @ 160): sum_ijkl M3[r,ijk] M3[s,ijl] M2[t,kl]
#pragma unroll
    for (int r = 0; r < 5; ++r)
#pragma unroll
        for (int s = 0; s <= r; ++s) {
            int pair = r * (r + 1) / 2 + s;
            float T[3][3];
#pragma unroll
            for (int k = 0; k < 3; ++k)
#pragma unroll
                for (int l = 0; l < 3; ++l) {
                    float acc = 0.0f;
#pragma unroll
                    for (int i = 0; i < 3; ++i)
#pragma unroll
                        for (int j = 0; j < 3; ++j)
                            acc += M3[r][s3i(i, j, k)] * M3[s][s3i(i, j, l)];
                    T[k][l] = acc;
                }
#pragma unroll
            for (int t = 0; t < 5; ++t) {
                float acc = 0.0f;
#pragma unroll
                for (int k = 0; k < 3; ++k)
#pragma unroll
                    for (int l = 0; l < 3; ++l) acc += T[k][l] * M2[t][s2i(k, l)];
                o[160 + pair * 5 + t] = acc;
            }
        }
    // c7 (125 feats @ 235): sum_ijk M3[r,ijk] M2[s,ij] M1[t,k]
#pragma unroll
    for (int r = 0; r < 5; ++r)
#pragma unroll
        for (int s = 0; s < 5; ++s) {
            float P[3];
#pragma unroll
            for (int k = 0; k < 3; ++k) {
                float acc = 0.0f;
#pragma unroll
                for (int i = 0; i < 3; ++i)
#pragma unroll
                    for (int j = 0; j < 3; ++j) acc += M3[r][s3i(i, j, k)] * M2[s][s2i(i, j)];
                P[k] = acc;
            }
#pragma unroll
            for (int t = 0; t < 5; ++t) {
                float acc = 0.0f;
#pragma unroll
                for (int k = 0; k < 3; ++k) acc += P[k] * M1[t][k];
                o[235 + r * 25 + s * 5 + t] = acc;
            }
        }
}

// ---- stage 2b helper: load a lane's A-operand slices with ALL indices,
// offsets and validity known at compile time (HI = upper half-wave).
// Invalid (K-padding) slots are compile-time zeros: no load issued.
template<bool HI>
__device__ __forceinline__ void gm_load_rows(const float* __restrict__ mp, float sc, int rr,
                                             float* a1, float* a2, float* a3)
{
    // c1: K=3 pad 4. lo: k=0,1  hi: k=2,(3=pad)
    a1[0] = sc * mp[5 + rr * 3 + (HI ? 2 : 0)];
    a1[1] = HI ? 0.0f : sc * mp[5 + rr * 3 + 1];
    // c2: K=9 pad 12
#pragma unroll
    for (int kk = 0; kk < 3; ++kk) {
        const int k0 = kk * 4 + (HI ? 2 : 0);
        const int k1 = k0 + 1;
        a2[kk * 2 + 0] = (k0 < 9) ? sc * mp[20 + rr * 6 + u2map(k0 < 9 ? k0 : 8)] : 0.0f;
        a2[kk * 2 + 1] = (k1 < 9) ? sc * mp[20 + rr * 6 + u2map(k1 < 9 ? k1 : 8)] : 0.0f;
    }
    // c3: K=27 pad 28
#pragma unroll
    for (int kk = 0; kk < 7; ++kk) {
        const int k0 = kk * 4 + (HI ? 2 : 0);
        const int k1 = k0 + 1;
        a3[kk * 2 + 0] = (k0 < 27) ? sc * mp[50 + rr * 10 + u3map(k0 < 27 ? k0 : 26)] : 0.0f;
        a3[kk * 2 + 1] = (k1 < 27) ? sc * mp[50 + rr * 10 + u3map(k1 < 27 ? k1 : 26)] : 0.0f;
    }
}

// ---- stage 2b: c1/c2/c3 Gram matrices via V_WMMA_F32_16X16X4_F32 -----------
// 4 waves / block, each wave = 3 atoms packed into the 16-row dim; D = A * A^T
// (same regs for A and B: the f32 A-operand layout (lane=row, K in vgpr/half)
// transposes to the B-operand layout (lane=col, K in vgpr/half)).
// Loads happen in a hi/lo branch with pure immediate offsets; the wave
// reconverges before the WMMA sequence (EXEC all-ones at every v_wmma).
__global__ void __launch_bounds__(128) gm_gram_wmma_kernel(const float* __restrict__ M,
                                                           float* __restrict__ out, int nAtoms)
{
    int  lane = threadIdx.x & 31;
    int  wave = threadIdx.x >> 5;
    int  base = (blockIdx.x * 4 + wave) * 3;
    int  n16  = lane & 15;
    bool hi   = lane >= 16;
    int  trow = n16 / 5;
    int  rr   = n16 - 5 * trow;          // radial row 0..4
    bool rowValid = (n16 < 15);
    int  atom = base + trow;
    bool av   = rowValid && (atom < nAtoms);
    const float* mp = M + (size_t)(av ? atom : 0) * NCOMP;   // clamped: always in-bounds
    float sc = av ? 1.0f : 0.0f;

    float a1[2], a2[6], a3[14];
    if (hi) gm_load_rows<true >(mp, sc, rr, a1, a2, a3);
    else    gm_load_rows<false>(mp, sc, rr, a1, a2, a3);
    // (reconverged here: EXEC all ones for the WMMAs below)

    v8f acc1 = {0,0,0,0,0,0,0,0};
    v8f acc2 = {0,0,0,0,0,0,0,0};
    v8f acc3 = {0,0,0,0,0,0,0,0};

    {
        v2f aop; aop.x = a1[0]; aop.y = a1[1];
        acc1 = __builtin_amdgcn_wmma_f32_16x16x4_f32(false, aop, false, aop,
                                                     (short)0, acc1, false, false);
    }
#pragma unroll
    for (int kk = 0; kk < 3; ++kk) {
        v2f aop; aop.x = a2[kk * 2 + 0]; aop.y = a2[kk * 2 + 1];
        acc2 = __builtin_amdgcn_wmma_f32_16x16x4_f32(false, aop, false, aop,
                                                     (short)0, acc2, false, false);
    }
#pragma unroll
    for (int kk = 0; kk < 7; ++kk) {
        v2f aop; aop.x = a3[kk * 2 + 0]; aop.y = a3[kk * 2 + 1];
        acc3 = __builtin_amdgcn_wmma_f32_16x16x4_f32(false, aop, false, aop,
                                                     (short)0, acc3, false, false);
    }

    // Scatter lower-triangle of each atom's 5x5 diagonal block.
    // C/D layout: vgpr g, lane l -> row = g + (l<16 ? 0 : 8), col = l%16.
    int  colT = n16 / 5, j2 = n16 - 5 * colT;
    int  atomn = base + colT;
    if ((n16 < 15) && atomn < nAtoms) {
        float* o = out + (size_t)atomn * NFEAT;
#pragma unroll
        for (int g = 0; g < 8; ++g) {
            int rl = g, rh = g + 8;                       // both constant after unroll
            int rt_l = rl / 5, i2_l = rl - 5 * rt_l;
            int rt_h = rh / 5, i2_h = rh - 5 * rt_h;
            bool val = hi ? (rh < 15) : (rl < 15);
            int  rt  = hi ? rt_h : rt_l;
            int  i2  = hi ? i2_h : i2_l;
            if (val && rt == colT && i2 >= j2) {
                int pos = (i2 * (i2 + 1)) / 2 + j2;
                o[5  + pos] = acc1[g];
                o[20 + pos] = acc2[g];
                o[35 + pos] = acc3[g];
            }
        }
    }
}

// ---- launch ---------------------------------------------------------------
extern "C" void kernel_launch(void* const* d_in, const int* in_sizes, int n_in,
                              void* d_out, int out_size, void* d_ws, size_t ws_size,
                              hipStream_t stream)
{
    const float* R   = (const float*)d_in[0];
    const int*   Z   = (const int*)  d_in[1];
    const int*   idx = (const int*)  d_in[2];
    const float* emb = (const float*)d_in[3];
    float*       out = (float*)d_out;

    int nAtoms = in_sizes[1];
    int nEdges = in_sizes[2] / 2;
    float* M = (float*)d_ws;                    // nAtoms * 100 floats (~8 MB)

    hipMemsetAsync(M, 0, (size_t)nAtoms * NCOMP * sizeof(float), stream);

    gm_edge_kernel<<<(nEdges + 255) / 256, 256, 0, stream>>>(R, Z, idx, emb, M, nEdges);
    gm_atom_kernel<<<(nAtoms + 127) / 128, 128, 0, stream>>>(M, out, nAtoms);
    gm_gram_wmma_kernel<<<(nAtoms + 11) / 12, 128, 0, stream>>>(M, out, nAtoms);
}